// VectorQuantizer_87582973100707
// MI455X (gfx1250) — compile-verified
//
#include <hip/hip_runtime.h>

typedef __attribute__((ext_vector_type(2))) float v2f;
typedef __attribute__((ext_vector_type(8))) float v8f;

#define D_DIM    64
#define K_CODES  8192
#define HW_SZ    1024
#define DHW_SZ   65536
#define N_VEC    32768
#define N_ELEM   2097152      // B*D*H*W
#define LDS_PITCH 68          // padded row pitch (floats) for bank-conflict-free b64 reads
#define COLS_PER_ITER 32      // two 16-wide WMMA column tiles per iteration
#define ITERS (K_CODES / COLS_PER_ITER)   // 256
#define BETA     0.25f

// ---------------------------------------------------------------------------
// Kernel 0: c2[k] = ||codebook[k]||^2 ; also zero the loss accumulator.
// ---------------------------------------------------------------------------
__global__ __launch_bounds__(256) void vq_c2_init_kernel(const float* __restrict__ cb,
                                                         float* __restrict__ c2,
                                                         float* __restrict__ loss_acc) {
  int k = blockIdx.x * 256 + threadIdx.x;      // 8192 threads
  if (k == 0) *loss_acc = 0.0f;
  const float4* p = (const float4*)(cb + (size_t)k * D_DIM);
  float s = 0.0f;
#pragma unroll
  for (int i = 0; i < D_DIM / 4; ++i) {
    float4 v = p[i];
    s += v.x * v.x + v.y * v.y + v.z * v.z + v.w * v.w;
  }
  c2[k] = s;
}

// ---------------------------------------------------------------------------
// Kernel 1: fused GEMM + argmin via V_WMMA_F32_16X16X4_F32.
// Each wave: 16 rows of Z kept in regs as A (ISA 16x4 fp32 layout), scans all
// 8192 codes in 32-column steps (two independent 16x16 WMMA accumulator
// chains per step). Code tiles staged in double-buffered LDS, shared by the
// 8 waves of the WG. dist = ||c||^2 - 2 * z.c  (||z||^2 constant per row).
// ---------------------------------------------------------------------------
__global__ __launch_bounds__(256) void vq_argmin_kernel(const float* __restrict__ x,
                                                        const float* __restrict__ cb,
                                                        const float* __restrict__ c2,
                                                        int*   __restrict__ idx_ws,
                                                        float* __restrict__ idx_out) {
  __shared__ float lds[2][32 * LDS_PITCH];

  const int tid  = threadIdx.x;
  const int wave = tid >> 5;
  const int lane = tid & 31;
  const int half = lane >> 4;    // 0: K pair {0,1}, 1: K pair {2,3} (ISA A/B layout)
  const int l16  = lane & 15;
  const int row_base = blockIdx.x * 128 + wave * 16;

  // ---- Load A strip (16 rows x 64 D) into registers, ISA 16x4 fp32 layout ----
  // row n = (b*H + h)*W + w ; x element = x[b*DHW + d*HW + hw]
  v2f a[16];
  {
    const int n  = row_base + l16;
    const int b  = n >> 10;
    const int hw = n & (HW_SZ - 1);
    const float* xp = x + (size_t)b * DHW_SZ + hw;
#pragma unroll
    for (int s = 0; s < 16; ++s) {
      const int d = 4 * s + 2 * half;
      v2f av;
      av.x = xp[(size_t)d * HW_SZ];
      av.y = xp[(size_t)(d + 1) * HW_SZ];
      a[s] = av;
    }
  }

  float bestD0[8], bestD1[8];
  int   bestK0[8], bestK1[8];
#pragma unroll
  for (int r = 0; r < 8; ++r) {
    bestD0[r] = 3.4e38f; bestK0[r] = 0;
    bestD1[r] = 3.4e38f; bestK1[r] = 0;
  }

  for (int t = 0; t < ITERS; ++t) {
    const int col0 = t * COLS_PER_ITER;
    float* buf = lds[t & 1];

    // ---- stage 32 codes x 64 floats (8KB): two float4 per thread ----
    {
      const int flat = tid * 8;          // 0..2047
      const int code = flat >> 6;        // 0..31
      const int d    = flat & 63;
      const float4* src = (const float4*)(cb + (size_t)(col0 + code) * D_DIM + d);
      const float4 v0 = src[0];
      const float4 v1 = src[1];
      float* dst = buf + code * LDS_PITCH + d;
      dst[0] = v0.x; dst[1] = v0.y; dst[2] = v0.z; dst[3] = v0.w;
      dst[4] = v1.x; dst[5] = v1.y; dst[6] = v1.z; dst[7] = v1.w;
    }
    __syncthreads();   // double-buffered: one barrier per iteration suffices

    // ---- two independent 16-step WMMA accumulation chains ----
    v8f acc0 = {0.f, 0.f, 0.f, 0.f, 0.f, 0.f, 0.f, 0.f};
    v8f acc1 = {0.f, 0.f, 0.f, 0.f, 0.f, 0.f, 0.f, 0.f};
#pragma unroll
    for (int s = 0; s < 16; ++s) {
      const int doff = 4 * s + 2 * half;
      const float* bp0 = buf + l16 * LDS_PITCH + doff;          // codes col0+0..15
      const float* bp1 = buf + (16 + l16) * LDS_PITCH + doff;   // codes col0+16..31
      v2f bv0, bv1;
      bv0.x = bp0[0]; bv0.y = bp0[1];
      bv1.x = bp1[0]; bv1.y = bp1[1];
      acc0 = __builtin_amdgcn_wmma_f32_16x16x4_f32(
          false, a[s], false, bv0, (short)0, acc0, false, false);
      acc1 = __builtin_amdgcn_wmma_f32_16x16x4_f32(
          false, a[s], false, bv1, (short)0, acc1, false, false);
    }

    // ---- running argmin: lane's acc[r] is row M=r+8*half, col = tile col l16 ----
    const float cc0 = c2[col0 + l16];
    const float cc1 = c2[col0 + 16 + l16];
    const int   k0  = col0 + l16;
    const int   k1  = col0 + 16 + l16;
#pragma unroll
    for (int r = 0; r < 8; ++r) {
      const float dist0 = cc0 - 2.0f * acc0[r];
      const float dist1 = cc1 - 2.0f * acc1[r];
      if (dist0 < bestD0[r]) { bestD0[r] = dist0; bestK0[r] = k0; }
      if (dist1 < bestD1[r]) { bestD1[r] = dist1; bestK1[r] = k1; }
    }
  }

  // ---- merge the two column streams, then reduce across the 16-lane half ----
#pragma unroll
  for (int r = 0; r < 8; ++r) {
    float d = bestD0[r];
    int   k = bestK0[r];
    if (bestD1[r] < d || (bestD1[r] == d && bestK1[r] < k)) { d = bestD1[r]; k = bestK1[r]; }
#pragma unroll
    for (int m = 8; m >= 1; m >>= 1) {
      const float od = __shfl_xor(d, m, 32);
      const int   ok = __shfl_xor(k, m, 32);
      if (od < d || (od == d && ok < k)) { d = od; k = ok; }
    }
    if (l16 == 0) {
      const int n = row_base + 8 * half + r;
      idx_ws[n]  = k;
      idx_out[n] = (float)k;   // encoding_indices in output buffer
    }
  }
}

// ---------------------------------------------------------------------------
// Kernel 2: gather z_q (straight-through output == z_q), accumulate SSE.
// ---------------------------------------------------------------------------
__global__ __launch_bounds__(256) void vq_gather_kernel(const float* __restrict__ x,
                                                        const float* __restrict__ cb,
                                                        const int* __restrict__ idx,
                                                        float* __restrict__ zq_out,
                                                        float* __restrict__ loss_acc) {
  const int gid = blockIdx.x * 256 + threadIdx.x;   // over B*D*H*W, output order
  const int b  = gid >> 16;
  const int d  = (gid >> 10) & 63;
  const int hw = gid & (HW_SZ - 1);
  const int n  = (b << 10) + hw;
  const int k  = idx[n];
  const float v  = cb[(size_t)k * D_DIM + d];
  const float xv = x[gid];
  zq_out[gid] = v;                                  // z_q_st == z_q numerically
  const float diff = v - xv;
  float s = diff * diff;
#pragma unroll
  for (int m = 16; m >= 1; m >>= 1) s += __shfl_xor(s, m, 32);
  if ((threadIdx.x & 31) == 0) atomicAdd(loss_acc, s);
}

// ---------------------------------------------------------------------------
// Kernel 3: vq_loss = (1 + beta) * SSE / (B*D*H*W)
// ---------------------------------------------------------------------------
__global__ void vq_finalize_kernel(const float* __restrict__ loss_acc,
                                   float* __restrict__ loss_out) {
  *loss_out = (1.0f + BETA) * (*loss_acc) / (float)N_ELEM;
}

// ---------------------------------------------------------------------------
extern "C" void kernel_launch(void* const* d_in, const int* in_sizes, int n_in,
                              void* d_out, int out_size, void* d_ws, size_t ws_size,
                              hipStream_t stream) {
  const float* x  = (const float*)d_in[0];   // (32,64,32,32) fp32
  const float* cb = (const float*)d_in[1];   // (8192,64) fp32

  // workspace layout
  float* c2       = (float*)d_ws;                                   // 8192 f32
  int*   idx_ws   = (int*)((char*)d_ws + (size_t)K_CODES * 4);      // 32768 i32
  float* loss_acc = (float*)((char*)d_ws + (size_t)K_CODES * 4 + (size_t)N_VEC * 4);

  // output layout: [z_q_st (2097152)] [vq_loss (1)] [indices (32768)]
  float* out      = (float*)d_out;
  float* zq_out   = out;
  float* loss_out = out + N_ELEM;
  float* idx_out  = out + N_ELEM + 1;

  vq_c2_init_kernel<<<K_CODES / 256, 256, 0, stream>>>(cb, c2, loss_acc);
  vq_argmin_kernel<<<N_VEC / 128, 256, 0, stream>>>(x, cb, c2, idx_ws, idx_out);
  vq_gather_kernel<<<N_ELEM / 256, 256, 0, stream>>>(x, cb, idx_ws, zq_out, loss_acc);
  vq_finalize_kernel<<<1, 1, 0, stream>>>(loss_acc, loss_out);
}